// MixedActivationLayer_71846212927677
// MI455X (gfx1250) — compile-verified
//
#include <hip/hip_runtime.h>

// MixedActivationLayer for MI455X (gfx1250).
//
// Shape: x f32[64, 256, 56, 56], NCHW. sel[c] = (c/16) % 3 -> relu/tanh/sigmoid.
// Pure streaming elementwise op: 205.5 MB in + 205.5 MB out = 411 MB traffic.
// HBM roofline @ 23.3 TB/s => ~17.6 us floor; arithmetic (<=1 transcendental
// per 8 bytes) is free. WMMA is inapplicable (no contraction); the gfx1250
// levers used instead:
//   - wave32 + b128 vector loads/stores (3136 f32 per channel = 784 float4s,
//     so a float4 never crosses a channel boundary -> uniform selector)
//   - non-temporal hints (411 MB stream > 192 MB L2, zero reuse)
//   - v_tanh_f32 (new CDNA5 hardware transcendental) for both tanh and
//     sigmoid channels (sigmoid(x) = 0.5*tanh(0.5x)+0.5 -> 1 trans op)

typedef float v4f __attribute__((ext_vector_type(4)));

#define C_CHANNELS 256u
#define HW_ELEMS   3136u   // 56*56
#define HW_VEC4    784u    // 3136/4

__device__ __forceinline__ float tanh_hw(float x) {
#if __has_builtin(__builtin_amdgcn_tanhf)
  return __builtin_amdgcn_tanhf(x);
#else
  float r;
  // CDNA5 V_TANH_F32. TRANS-op hazard rule (ISA 7.4.3): one independent op /
  // V_NOP before the destination may be consumed -- compiler cannot see
  // through inline asm, so pad explicitly.
  asm("v_tanh_f32 %0, %1\n\tv_nop" : "=v"(r) : "v"(x));
  return r;
#endif
}

__device__ __forceinline__ float sigmoid_hw(float x) {
  // sigmoid(x) = 0.5*tanh(0.5*x) + 0.5  (exact identity)
  return fmaf(tanh_hw(0.5f * x), 0.5f, 0.5f);
}

__global__ __launch_bounds__(256) void mixed_act_vec4_kernel(
    const float* __restrict__ x, float* __restrict__ y, unsigned nvec) {
  unsigned i = (unsigned)blockIdx.x * blockDim.x + threadIdx.x;
  if (i >= nvec) return;

  // vec4 index -> channel: c = (i / 784) % 256 ; selector = (c/16) % 3
  unsigned c   = (i / HW_VEC4) & (C_CHANNELS - 1u);
  unsigned sel = (c >> 4) % 3u;

  const v4f* __restrict__ xv = (const v4f*)x;
  v4f*       __restrict__ yv = (v4f*)y;

  v4f v = __builtin_nontemporal_load(xv + i);  // global_load_b128 th:NT
  v4f r;
  if (sel == 0u) {            // relu
    r.x = fmaxf(v.x, 0.0f);
    r.y = fmaxf(v.y, 0.0f);
    r.z = fmaxf(v.z, 0.0f);
    r.w = fmaxf(v.w, 0.0f);
  } else if (sel == 1u) {     // tanh
    r.x = tanh_hw(v.x);
    r.y = tanh_hw(v.y);
    r.z = tanh_hw(v.z);
    r.w = tanh_hw(v.w);
  } else {                    // sigmoid
    r.x = sigmoid_hw(v.x);
    r.y = sigmoid_hw(v.y);
    r.z = sigmoid_hw(v.z);
    r.w = sigmoid_hw(v.w);
  }
  __builtin_nontemporal_store(r, yv + i);      // global_store_b128 th:NT
}

// Generic scalar tail (unused for this shape: 51,380,224 % 4 == 0).
__global__ void mixed_act_tail_kernel(const float* __restrict__ x,
                                      float* __restrict__ y,
                                      unsigned start, unsigned n) {
  unsigned i = start + blockIdx.x * blockDim.x + threadIdx.x;
  if (i >= n) return;
  unsigned c   = (i / HW_ELEMS) & (C_CHANNELS - 1u);
  unsigned sel = (c >> 4) % 3u;
  float v = x[i];
  float r;
  if (sel == 0u)      r = fmaxf(v, 0.0f);
  else if (sel == 1u) r = tanh_hw(v);
  else                r = sigmoid_hw(v);
  y[i] = r;
}

extern "C" void kernel_launch(void* const* d_in, const int* in_sizes, int n_in,
                              void* d_out, int out_size, void* d_ws, size_t ws_size,
                              hipStream_t stream) {
  (void)in_sizes; (void)n_in; (void)d_ws; (void)ws_size;
  const float* x = (const float*)d_in[0];
  float* y = (float*)d_out;

  unsigned n    = (unsigned)out_size;
  unsigned nvec = n / 4u;

  if (nvec) {
    unsigned blocks = (nvec + 255u) / 256u;
    mixed_act_vec4_kernel<<<blocks, 256, 0, stream>>>(x, y, nvec);
  }
  unsigned tail_start = nvec * 4u;
  if (tail_start < n) {
    unsigned tail = n - tail_start;
    mixed_act_tail_kernel<<<(tail + 63u) / 64u, 64, 0, stream>>>(x, y, tail_start, n);
  }
}